// Slot_Merger_21646635172285
// MI455X (gfx1250) — compile-verified
//
#include <hip/hip_runtime.h>

typedef __attribute__((ext_vector_type(2))) float v2f;
typedef __attribute__((ext_vector_type(8))) float v8f;

#define B_  256
#define S_  24
#define D_  256
#define N_  8192
#define EPSF 1e-8f

// ---------------------------------------------------------------------------
// Kernel A1: rowsum[b,j] = sum_n patch[b,j,n].
// One block per (b,j) row -> 6144 blocks streaming 192 MiB with b128 loads.
// ---------------------------------------------------------------------------
__global__ void sm_rowsum_kernel(const float* __restrict__ patch,
                                 float* __restrict__ rowsum)
{
    const int row = blockIdx.x;                  // flat b*S + j
    const int t   = threadIdx.x;
    const float4* r4 = (const float4*)(patch + (size_t)row * N_);

    float acc = 0.0f;
    #pragma unroll
    for (int n = t; n < N_ / 4; n += 256) {
        float4 v = r4[n];
        acc += (v.x + v.y) + (v.z + v.w);
    }

    __shared__ float red[256];
    red[t] = acc;
    __syncthreads();
    for (int off = 128; off > 0; off >>= 1) {
        if (t < off) red[t] += red[t + off];
        __syncthreads();
    }
    if (t == 0) rowsum[row] = red[0];
}

// ---------------------------------------------------------------------------
// Kernel A2: per-batch scales (tiny).
//   inv_counts[b,i] = 1/(counts+eps)
//   inv_denom[b,i]  = 1/( sum_{j: cl[j]==i} rowsum[b,j] + N*eps )
//   slot_nums[b]    = #{i : counts[i] > 0}   (float, d_out tail)
// ---------------------------------------------------------------------------
__global__ void sm_scales_kernel(const int*   __restrict__ clusters,
                                 const float* __restrict__ rowsum,
                                 float* __restrict__ inv_counts,
                                 float* __restrict__ inv_denom,
                                 float* __restrict__ slot_nums_out)
{
    const int b = blockIdx.x;
    const int t = threadIdx.x;                   // blockDim = 32 (one wave)

    __shared__ int   s_cl[S_];
    __shared__ float s_cnt[S_];
    if (t < S_) { s_cl[t] = clusters[b * S_ + t]; s_cnt[t] = 0.0f; }
    __syncthreads();
    if (t < S_) atomicAdd(&s_cnt[s_cl[t]], 1.0f);
    __syncthreads();

    if (t < S_) {
        float d = (float)N_ * EPSF;
        for (int j = 0; j < S_; ++j)
            if (s_cl[j] == t) d += rowsum[b * S_ + j];
        inv_denom[b * S_ + t]  = 1.0f / d;
        inv_counts[b * S_ + t] = 1.0f / (s_cnt[t] + EPSF);
    }
    if (t == 0) {
        float nn = 0.0f;
        for (int i = 0; i < S_; ++i) nn += (s_cnt[i] > 0.0f) ? 1.0f : 0.0f;
        slot_nums_out[b] = nn;
    }
}

// ---------------------------------------------------------------------------
// Kernel B: fused  out[b] = (C[b] @ X[b] + eps?) * scale[row]  via f32 WMMA.
//   COLS is a template constant so all row strides are 24-bit immediate
//   offsets; the per-lane halfSel row skew is folded into base pointers
//   once, outside the streaming loop. A fragments & scales are hoisted.
// ---------------------------------------------------------------------------
template <int COLS, bool ADD_EPS>
__global__ void sm_merge_wmma_kernel(const float* __restrict__ X,
                                     const int*   __restrict__ clusters,
                                     const float* __restrict__ rowScale,
                                     float* __restrict__ out)
{
    const int b = blockIdx.x;
    const int t = threadIdx.x;

    __shared__ int   s_cl[S_];
    __shared__ float s_scale[S_];
    if (t < S_) {
        s_cl[t]    = clusters[b * S_ + t];
        s_scale[t] = rowScale[b * S_ + t];
    }
    __syncthreads();

    const int lane    = t & 31;
    const int wave    = t >> 5;
    const int halfSel = lane >> 4;   // 0: lanes 0-15, 1: lanes 16-31
    const int lrow    = lane & 15;

    // A-matrix 16x4 f32 fragments: lanes 0-15 hold {K=0,1}, lanes 16-31 {K=2,3}.
    v2f afrag[2][6];
    #pragma unroll
    for (int m = 0; m < 2; ++m) {
        const int row = m * 16 + lrow;           // rows 24..31 never match
        #pragma unroll
        for (int k = 0; k < 6; ++k) {
            const int j0 = k * 4 + halfSel * 2;
            v2f a;
            a.x = (s_cl[j0]     == row) ? 1.0f : 0.0f;
            a.y = (s_cl[j0 + 1] == row) ? 1.0f : 0.0f;
            afrag[m][k] = a;
        }
    }

    // Hoist per-lane output scales out of the streaming loop.
    float scale0[8], scale1[8];
    #pragma unroll
    for (int r = 0; r < 8; ++r) {
        scale0[r] = s_scale[halfSel * 8 + r];                      // rows 0..15
        scale1[r] = (halfSel == 0) ? s_scale[16 + r] : 0.0f;       // rows 16..23
    }
    const bool storeHi = (halfSel == 0);
    const float epsf = ADD_EPS ? EPSF : 0.0f;

    const int nTiles  = COLS >> 4;
    const int strideT = gridDim.y * 8;
    const int tile0   = blockIdx.y * 8 + wave;
    const size_t step = (size_t)strideT * 16;

    const size_t batchOfs = (size_t)b * S_ * COLS;
    const int    col0     = tile0 * 16 + lrow;
    // Base pointers with per-lane row skew folded in once.
    const float* Bp  = X   + batchOfs + (size_t)(halfSel * 2) * COLS + col0;
    float*       Op0 = out + batchOfs + (size_t)(halfSel * 8) * COLS + col0;
    float*       Op1 = out + batchOfs + (size_t)(16 + halfSel * 8) * COLS + col0;

    for (int tile = tile0; tile < nTiles;
         tile += strideT, Bp += step, Op0 += step, Op1 += step) {
        v8f acc0 = {};
        v8f acc1 = {};
        #pragma unroll
        for (int k = 0; k < 6; ++k) {
            v2f bf;                               // rows k*4+halfSel*2 (+1)
            bf.x = Bp[(size_t)(k * 4) * COLS];
            bf.y = Bp[(size_t)(k * 4 + 1) * COLS];
            acc0 = __builtin_amdgcn_wmma_f32_16x16x4_f32(
                       false, afrag[0][k], false, bf, (short)0, acc0, false, false);
            acc1 = __builtin_amdgcn_wmma_f32_16x16x4_f32(
                       false, afrag[1][k], false, bf, (short)0, acc1, false, false);
        }
        // D layout: VGPR r -> row (r + halfSel*8), col = lane&15.
        #pragma unroll
        for (int r = 0; r < 8; ++r)
            Op0[(size_t)r * COLS] = (acc0[r] + epsf) * scale0[r];
        if (storeHi) {
            #pragma unroll
            for (int r = 0; r < 8; ++r)
                Op1[(size_t)r * COLS] = (acc1[r] + epsf) * scale1[r];
        }
    }
}

// ---------------------------------------------------------------------------
extern "C" void kernel_launch(void* const* d_in, const int* in_sizes, int n_in,
                              void* d_out, int out_size, void* d_ws, size_t ws_size,
                              hipStream_t stream)
{
    (void)in_sizes; (void)n_in; (void)out_size; (void)ws_size;

    const float* slots    = (const float*)d_in[0];
    const float* patch    = (const float*)d_in[1];
    const int*   clusters = (const int*)  d_in[2];

    float* out       = (float*)d_out;
    float* out_slots = out;
    float* out_patch = out + (size_t)B_ * S_ * D_;
    float* out_nums  = out_patch + (size_t)B_ * S_ * N_;

    float* inv_counts = (float*)d_ws;            // B*S floats
    float* inv_denom  = inv_counts + B_ * S_;    // B*S floats
    float* rowsum     = inv_denom  + B_ * S_;    // B*S floats

    // Pass 1: row sums (6144 blocks, streams patch once) + per-batch scales.
    hipLaunchKernelGGL(sm_rowsum_kernel, dim3(B_ * S_), dim3(256), 0, stream,
                       patch, rowsum);
    hipLaunchKernelGGL(sm_scales_kernel, dim3(B_), dim3(32), 0, stream,
                       clusters, rowsum, inv_counts, inv_denom, out_nums);

    // Pass 2a: new_slots = (C @ slots) * inv_counts   (COLS = 256)
    hipLaunchKernelGGL((sm_merge_wmma_kernel<D_, false>), dim3(B_, 2), dim3(256),
                       0, stream, slots, clusters, inv_counts, out_slots);

    // Pass 2b: new_patch_attn = (C @ patch + eps) * inv_denom   (COLS = 8192)
    hipLaunchKernelGGL((sm_merge_wmma_kernel<N_, true>), dim3(B_, 8), dim3(256),
                       0, stream, patch, clusters, inv_denom, out_patch);
}